// Model_23880018165847
// MI455X (gfx1250) — compile-verified
//
#include <hip/hip_runtime.h>
#include <float.h>

// Ball-query + masked channel max-pool for MI455X (gfx1250, wave32).
// Phase 1: V_WMMA_F32_16X16X4_F32 computes 16x16 query.point dot products
//          (K=3 padded to 4 with zeros), software-pipelined point loads.
// Phase 2: wave-uniform ballot -> sparse scalar bit-scan -> coalesced
//          feature-row max with lane==channel; LDS cross-wave reduction.

typedef __attribute__((ext_vector_type(2))) float v2f;
typedef __attribute__((ext_vector_type(8))) float v8f;

#define B_  2
#define P_  4096
#define Q_  512
#define C_  32
#define RADIUS_SQ 0.0144f
#define QTILE 16
#define PTILE 16
#define WAVES 8
#define NQT (Q_ / QTILE)   // 32 query tiles per batch
#define NPT (P_ / PTILE)   // 256 point tiles
#define ITERS (NPT / WAVES) // 32 exact iterations per wave

// Order-preserving float->uint mapping so LDS integer atomicMax implements float max.
__device__ __forceinline__ unsigned f2mono(float f) {
    unsigned u = __float_as_uint(f);
    return (u & 0x80000000u) ? ~u : (u | 0x80000000u);
}
__device__ __forceinline__ float mono2f(unsigned u) {
    return (u & 0x80000000u) ? __uint_as_float(u & 0x7fffffffu)
                             : __uint_as_float(~u);
}

__global__ __launch_bounds__(256)
void ball_query_maxpool(const float* __restrict__ points,
                        const float* __restrict__ queries,
                        const float* __restrict__ features,
                        float* __restrict__ out) {
    __shared__ unsigned pooled[QTILE][C_];   // monotone-encoded running max
    __shared__ unsigned anymask;             // bit m: query m has >=1 neighbor

    const int tid  = threadIdx.x;
    const int lane = tid & 31;
    const int wave = tid >> 5;
    const int b    = blockIdx.x / NQT;
    const int qt   = blockIdx.x % NQT;

    // ---- init LDS reduction buffers ----
    const unsigned NEG = f2mono(-FLT_MAX);
    for (int i = tid; i < QTILE * C_; i += 256)
        ((unsigned*)pooled)[i] = NEG;
    if (tid == 0) anymask = 0u;
    __syncthreads();

    const float* ptsB  = points   + (size_t)b * P_ * 3;
    const float* qryB  = queries  + ((size_t)b * Q_ + (size_t)qt * QTILE) * 3;
    const float* featB = features + (size_t)b * P_ * C_;

    const int half = lane >> 4;     // 0: K=0,1 / M rows 0-7 ; 1: K=2,3 / M rows 8-15
    const int mloc = lane & 15;

    // ---- A operand: 16x4 queries (x,y | z,0 split across lane halves) ----
    float qx = qryB[mloc * 3 + 0];
    float qy = qryB[mloc * 3 + 1];
    float qz = qryB[mloc * 3 + 2];
    v2f A;
    A.x = half ? qz   : qx;
    A.y = half ? 0.0f : qy;

    // |q|^2 for the 8 M-rows this lane's half of the C/D layout covers
    float qq[8];
#pragma unroll
    for (int v = 0; v < 8; ++v) {
        int mq = v + 8 * half;
        float x = qryB[mq * 3 + 0], y = qryB[mq * 3 + 1], z = qryB[mq * 3 + 2];
        qq[v] = x * x + y * y + z * z;
    }

    float accLo[8], accHi[8];        // rows v (lanes' low half) and v+8
#pragma unroll
    for (int v = 0; v < 8; ++v) { accLo[v] = -FLT_MAX; accHi[v] = -FLT_MAX; }
    unsigned any = 0u;               // wave-uniform (derived from ballots)

    // ---- software-pipelined point loop: 8 waves stripe 256 tiles ----
    int pt = wave;
    float px = ptsB[(size_t)(pt * PTILE + mloc) * 3 + 0];
    float py = ptsB[(size_t)(pt * PTILE + mloc) * 3 + 1];
    float pz = ptsB[(size_t)(pt * PTILE + mloc) * 3 + 2];

    for (int iter = 0; iter < ITERS; ++iter) {
        // issue next tile's load NOW; its s_wait lands a full iteration later
        const int ptn = pt + WAVES;
        const int ptc = (ptn < NPT) ? ptn : pt;          // clamp (uniform, no branch)
        const float nx = ptsB[(size_t)(ptc * PTILE + mloc) * 3 + 0];
        const float ny = ptsB[(size_t)(ptc * PTILE + mloc) * 3 + 1];
        const float nz = ptsB[(size_t)(ptc * PTILE + mloc) * 3 + 2];
        __builtin_prefetch(ptsB + (size_t)((pt + 2 * WAVES < NPT ? pt + 2 * WAVES
                                                                 : pt) * PTILE) * 3,
                           0, 3);                        // keep L0 warm 2 tiles ahead

        const int pbase = pt * PTILE;
        v2f Bv;                      // B operand: 4x16 points
        Bv.x = half ? pz   : px;
        Bv.y = half ? 0.0f : py;
        const float pp = px * px + py * py + pz * pz;

        v8f c = {};
        // 16x16 dot products: c[v] = dot(query_{v+8*half}, point_{lane&15})
        c = __builtin_amdgcn_wmma_f32_16x16x4_f32(
                /*neg_a=*/false, A, /*neg_b=*/false, Bv,
                /*c_mod=*/(short)0, c, /*reuse_a=*/false, /*reuse_b=*/false);

#pragma unroll
        for (int v = 0; v < 8; ++v) {
            float d2 = qq[v] + pp - 2.0f * c[v];
            unsigned full = __builtin_amdgcn_ballot_w32(d2 <= RADIUS_SQ);
            unsigned mlo = full & 0xFFFFu;   // row m=v,   cols = lanes 0-15
            unsigned mhi = full >> 16;       // row m=v+8, cols = lanes 16-31
            if (mlo) any |= 1u << v;
            while (mlo) {                    // sparse scalar bit-scan (uniform)
                int nn = __builtin_ctz(mlo); mlo &= mlo - 1;
                accLo[v] = fmaxf(accLo[v],
                                 featB[(size_t)(pbase + nn) * C_ + lane]);
            }
            if (mhi) any |= 1u << (v + 8);
            while (mhi) {
                int nn = __builtin_ctz(mhi); mhi &= mhi - 1;
                accHi[v] = fmaxf(accHi[v],
                                 featB[(size_t)(pbase + nn) * C_ + lane]);
            }
        }

        px = nx; py = ny; pz = nz; pt = ptn;   // rotate pipeline registers
    }

    // ---- cross-wave reduction in LDS (lane == channel) ----
#pragma unroll
    for (int v = 0; v < 8; ++v) {
        atomicMax(&pooled[v][lane],     f2mono(accLo[v]));
        atomicMax(&pooled[v + 8][lane], f2mono(accHi[v]));
    }
    if (lane == 0) atomicOr(&anymask, any);
    __syncthreads();

    // ---- write (B,Q,C); empty queries emit zeros ----
    float* outB = out + ((size_t)b * Q_ + (size_t)qt * QTILE) * C_;
    unsigned am = anymask;
    for (int i = tid; i < QTILE * C_; i += 256) {
        int m = i >> 5, ch = i & 31;
        outB[i] = ((am >> m) & 1u) ? mono2f(pooled[m][ch]) : 0.0f;
    }
}

extern "C" void kernel_launch(void* const* d_in, const int* in_sizes, int n_in,
                              void* d_out, int out_size, void* d_ws, size_t ws_size,
                              hipStream_t stream) {
    const float* points   = (const float*)d_in[0];
    const float* queries  = (const float*)d_in[1];
    const float* features = (const float*)d_in[2];
    float* out = (float*)d_out;
    dim3 grid(B_ * NQT);   // 64 blocks x 8 wave32s = 512 waves
    ball_query_maxpool<<<grid, 256, 0, stream>>>(points, queries, features, out);
}